// TrialityLoss_25838523252882
// MI455X (gfx1250) — compile-verified
//
#include <hip/hip_runtime.h>
#include <math.h>

// CDNA5 / gfx1250, wave32. fp32 WMMA: V_WMMA_F32_16X16X4_F32.
typedef __attribute__((ext_vector_type(2))) float v2f;
typedef __attribute__((ext_vector_type(8))) float v8f;

#define WAVES_PER_BLOCK 8
#define ST 20                 // padded LDS row stride in floats (multiple of 4 -> 16B-aligned rows)
#define MAT_FLOATS (16 * ST)  // one 16x16 tile buffer

// ---- fragment helpers (wave32 WMMA fp32 layouts) -------------------------
// A 16x4 f32: lane l holds row m=l%16, K = 2*(l/16)+{0,1}  (2 VGPRs)
// B 4x16 f32: lane l holds col n=l%16, K = 2*(l/16)+{0,1}  (2 VGPRs)
// C 16x16 f32: lane l, vgpr r -> (m = r + 8*(l/16), n = l%16) (8 VGPRs)

__device__ __forceinline__ v8f mm16_lds(const float* X, const float* Y, int lane, bool transposeA) {
  v8f c = {};
  const int mn = lane & 15;
  const int kp = (lane >> 4) << 1;
#pragma unroll
  for (int kc = 0; kc < 4; ++kc) {
    const int ks = kc * 4 + kp;
    v2f a;
    if (transposeA) {             // A-fragment of X^T read straight from LDS
      a.x = X[ks * ST + mn];
      a.y = X[(ks + 1) * ST + mn];
    } else {
      a = *(const v2f*)(X + mn * ST + ks);
    }
    v2f b;
    b.x = Y[ks * ST + mn];
    b.y = Y[(ks + 1) * ST + mn];
    c = __builtin_amdgcn_wmma_f32_16x16x4_f32(false, a, false, b, (short)0, c, false, false);
  }
  return c;
}

__device__ __forceinline__ v8f mm16_cachedA(const v2f Ba[4], const float* Y, int lane) {
  v8f c = {};
  const int n  = lane & 15;
  const int kp = (lane >> 4) << 1;
#pragma unroll
  for (int kc = 0; kc < 4; ++kc) {
    const int ks = kc * 4 + kp;
    v2f b;
    b.x = Y[ks * ST + n];
    b.y = Y[(ks + 1) * ST + n];
    c = __builtin_amdgcn_wmma_f32_16x16x4_f32(false, Ba[kc], false, b, (short)0, c, false, false);
  }
  return c;
}

__device__ __forceinline__ void storeC(float* X, const v8f& c, int lane) {
  const int n  = lane & 15;
  const int mb = (lane >> 4) << 3;
#pragma unroll
  for (int r = 0; r < 8; ++r) X[(mb + r) * ST + n] = c[r];
}

// ---- main kernel: one wave handles 2 of the 8x8 blocks as a block-diagonal 16x16 tile
__global__ __launch_bounds__(256) void TrialityLoss_expm_ortho(const float* __restrict__ P,
                                                               float* __restrict__ partial) {
  __shared__ float lds[WAVES_PER_BLOCK * 2 * MAT_FLOATS];
  __shared__ float wsum[WAVES_PER_BLOCK];
  const int tid  = threadIdx.x;
  const int wave = tid >> 5;
  const int lane = tid & 31;

  // zero off-diagonal staging area (whole LDS for simplicity)
  for (int i = tid; i < WAVES_PER_BLOCK * 2 * MAT_FLOATS; i += 256) lds[i] = 0.0f;
  __syncthreads();

  float* buf0 = lds + wave * 2 * MAT_FLOATS;
  float* buf1 = buf0 + MAT_FLOATS;

  // --- stage: 2 consecutive 8x8 fp32 matrices -> block-diagonal 16x16 tile (buf0)
  // wave reads 512 contiguous bytes: one global_load_b128 per lane
  const long long tile = (long long)blockIdx.x * WAVES_PER_BLOCK + wave;
  const float4 v = *(const float4*)(P + tile * 128 + lane * 4);
  {
    const int j  = lane * 4;       // element index within the 128 staged floats
    const int mi = j >> 6;         // which of the two matrices
    const int r  = (j & 63) >> 3;  // row in 8x8
    const int cb = j & 7;          // col base (0 or 4)
    *(float4*)(buf0 + (r + 8 * mi) * ST + cb + 8 * mi) = v;
  }
  __syncthreads();

  const int n  = lane & 15;
  const int mb = (lane >> 4) << 3;

  // --- B = (P - P^T) * (0.5 / 2^6): skew part pre-scaled for 6 squarings -> buf1
#pragma unroll
  for (int r = 0; r < 8; ++r) {
    const int m = mb + r;
    buf1[m * ST + n] = (buf0[m * ST + n] - buf0[n * ST + m]) * (1.0f / 128.0f);
  }
  __syncthreads();

  // cache A-side fragments of B in registers for the whole Horner phase
  v2f Ba[4];
  {
    const int m  = lane & 15;
    const int kp = (lane >> 4) << 1;
#pragma unroll
    for (int kc = 0; kc < 4; ++kc) Ba[kc] = *(const v2f*)(buf1 + m * ST + kc * 4 + kp);
  }

  // --- H = I + B/8 (start of order-8 Taylor Horner) -> buf0
#pragma unroll
  for (int r = 0; r < 8; ++r) {
    const int m = mb + r;
    buf0[m * ST + n] = buf1[m * ST + n] * (1.0f / 8.0f) + ((m == n) ? 1.0f : 0.0f);
  }
  __syncthreads();

  // --- Horner: H <- I + (B*H)/k for k = 7..1  (7 matmuls, ping-pong buffers)
  float* cur = buf0;
  float* nxt = buf1;
#pragma unroll
  for (int k = 7; k >= 1; --k) {
    v8f c = mm16_cachedA(Ba, cur, lane);
    const float inv = 1.0f / (float)k;
#pragma unroll
    for (int r = 0; r < 8; ++r) c[r] = c[r] * inv + ((mb + r == n) ? 1.0f : 0.0f);
    storeC(nxt, c, lane);
    float* t = cur; cur = nxt; nxt = t;
    __syncthreads();
  }

  // --- 6 repeated squarings: R <- R*R
#pragma unroll
  for (int sq = 0; sq < 6; ++sq) {
    v8f c = mm16_lds(cur, cur, lane, false);
    storeC(nxt, c, lane);
    float* t = cur; cur = nxt; nxt = t;
    __syncthreads();
  }

  // --- Gram: G = R^T * R (transposed A-fragment read directly from LDS)
  v8f g = mm16_lds(cur, cur, lane, true);

  // --- per-lane sum of squared deviation from I
  float ss = 0.0f;
#pragma unroll
  for (int r = 0; r < 8; ++r) {
    const float e = g[r] - ((mb + r == n) ? 1.0f : 0.0f);
    ss += e * e;
  }
  // reduce within each 16-lane half (lanes 0-15 -> matrix 0, 16-31 -> matrix 1;
  // off-diagonal-block entries are exactly zero by construction)
#pragma unroll
  for (int off = 1; off < 16; off <<= 1) ss += __shfl_xor(ss, off, 32);
  const float other = __shfl_xor(ss, 16, 32);
  if (lane == 0) wsum[wave] = sqrtf(ss) + sqrtf(other);
  __syncthreads();

  if (tid == 0) {
    float s = 0.0f;
#pragma unroll
    for (int w = 0; w < WAVES_PER_BLOCK; ++w) s += wsum[w];
    partial[blockIdx.x] = s;  // plain store: no pre-zeroing of d_ws needed
  }
}

// ---- deterministic final reduction: one block sums all per-block partials
__global__ __launch_bounds__(256) void TrialityLoss_reduce(const float* __restrict__ partial,
                                                           int nPartial, float invN,
                                                           float* __restrict__ out) {
  __shared__ float sm[256];
  float s = 0.0f;
  for (int i = threadIdx.x; i < nPartial; i += 256) s += partial[i];
  sm[threadIdx.x] = s;
  __syncthreads();
  for (int off = 128; off > 0; off >>= 1) {
    if ((int)threadIdx.x < off) sm[threadIdx.x] += sm[threadIdx.x + off];
    __syncthreads();
  }
  if (threadIdx.x == 0) out[0] = sm[0] * invN;
}

extern "C" void kernel_launch(void* const* d_in, const int* in_sizes, int n_in,
                              void* d_out, int out_size, void* d_ws, size_t ws_size,
                              hipStream_t stream) {
  const float* P = (const float*)d_in[0];
  float* out     = (float*)d_out;
  const long long nMat   = (long long)in_sizes[0] / 64;   // 524288 for the reference
  const int nTiles       = (int)(nMat / 2);               // 2 matrices per wave
  const int nBlocks      = nTiles / WAVES_PER_BLOCK;      // 16 matrices per block (N divisible)
  float* partial         = (float*)d_ws;                  // nBlocks floats of scratch

  TrialityLoss_expm_ortho<<<nBlocks, 256, 0, stream>>>(P, partial);
  TrialityLoss_reduce<<<1, 256, 0, stream>>>(partial, nBlocks, 1.0f / (float)nMat, out);
}